// Model_25881472926495
// MI455X (gfx1250) — compile-verified
//
#include <hip/hip_runtime.h>
#include <hip/hip_bf16.h>
#include <math.h>

typedef __attribute__((ext_vector_type(2))) float v2f;
typedef __attribute__((ext_vector_type(8))) float v8f;

#define NSTA  24
#define BATCH 32
#define ROWS  (BATCH * NSTA)   // 768
#define TILES (ROWS / 16)      // 48
#define XSTR  260              // 260 % 64 == 4 -> conflict-free A-row reads
#define HSTR  132              // 132 % 64 == 4

// ---------------------------------------------------------------------------
// Kernel 1: nearest-neighbor index + relative position per station.
// d[n,g] = (csta[n,0]-cera[i,j,0])^2 + (csta[n,1]-cera[i,j,1])^2, g = i*128+j
// argmin with first-occurrence tie-break (matches jnp.argmin).
// ---------------------------------------------------------------------------
__global__ __launch_bounds__(256) void nn_kernel(const float* __restrict__ csta,
                                                 const float* __restrict__ cera,
                                                 int* __restrict__ nn,
                                                 float* __restrict__ rel) {
  const int n = blockIdx.x;                       // 0..23
  const float s0 = csta[n * 2 + 0];
  const float s1 = csta[n * 2 + 1];
  float best = 3.4e38f;
  int bidx = 1 << 30;
  for (int g = threadIdx.x; g < 128 * 128; g += 256) {
    const int i = g >> 7, j = g & 127;
    const float c0 = cera[(i * 129 + j) * 2 + 0];
    const float c1 = cera[(i * 129 + j) * 2 + 1];
    const float d0 = s0 - c0, d1 = s1 - c1;
    const float d = d0 * d0 + d1 * d1;
    if (d < best || (d == best && g < bidx)) { best = d; bidx = g; }
  }
  __shared__ float sd[256];
  __shared__ int si[256];
  sd[threadIdx.x] = best;
  si[threadIdx.x] = bidx;
  __syncthreads();
  for (int off = 128; off > 0; off >>= 1) {
    if (threadIdx.x < off) {
      const float d2 = sd[threadIdx.x + off];
      const int i2 = si[threadIdx.x + off];
      if (d2 < sd[threadIdx.x] || (d2 == sd[threadIdx.x] && i2 < si[threadIdx.x])) {
        sd[threadIdx.x] = d2;
        si[threadIdx.x] = i2;
      }
    }
    __syncthreads();
  }
  if (threadIdx.x == 0) {
    const int g = si[0];
    nn[n] = g;
    const int i = g >> 7, j = g & 127;
    // pos = (csta[:,1], csta[:,0]); grid_pos = (cera[...,1], cera[...,0])
    rel[n * 2 + 0] = s1 - cera[(i * 129 + j) * 2 + 1];
    rel[n * 2 + 1] = s0 - cera[(i * 129 + j) * 2 + 0];
  }
}

// ---------------------------------------------------------------------------
// Fused forward kernel. One block per 16-row tile; 8 wave32 per block.
// All matmuls via V_WMMA_F32_16X16X4_F32 (fp32 A/B/C, K=4 per op).
// ---------------------------------------------------------------------------
struct Params {
  const float *obs, *era, *pan, *csta;
  const float *e1w, *e1b, *e2w, *e2b;
  const float *m1w[2], *m1b[2], *m2w[2], *m2b[2];
  const float *u1w[2], *u1b[2], *u2w[2], *u2b[2];
  const float *o1w, *o1b, *o2w, *o2b;
  const int* nn;
  const float* rel;
  float* out;
};

// One 16x16 output tile of  sOut = act(sIn[16 x inDim] @ W[inDim x ldw] + bias)
// A-frag (16x4 f32): lanes 0-15 -> K=k,k+1 ; lanes 16-31 -> K=k+2,k+3  (ISA 7.12.2)
// B-frag (4x16 f32): mirrored (VGPR0: K=k / k+2, VGPR1: K=k+1 / k+3)
__device__ __forceinline__ void mm_layer(const float* __restrict__ W,
                                         const float* __restrict__ bias,
                                         const float* __restrict__ sIn, int inDim, int inStride,
                                         float* __restrict__ sOut, int outStride,
                                         int ldw, bool act, int lane, int colBase) {
  const int half = lane >> 4;
  const int lr = lane & 15;
  const int col = colBase + lr;
  v8f acc = {};
  const int kEnd = (inDim + 3) & ~3;
  for (int k = 0; k < kEnd; k += 4) {
    const int ka = k + 2 * half;
    v2f a, bw;
    a.x = sIn[lr * inStride + ka];           // padded cols are zeroed in LDS
    a.y = sIn[lr * inStride + ka + 1];
    bw.x = (ka     < inDim) ? W[ka * ldw + col]       : 0.0f;
    bw.y = (ka + 1 < inDim) ? W[(ka + 1) * ldw + col] : 0.0f;
    acc = __builtin_amdgcn_wmma_f32_16x16x4_f32(false, a, false, bw,
                                                (short)0, acc, false, false);
  }
  const float bv = bias[col];
#pragma unroll
  for (int i = 0; i < 8; ++i) {
    float v = acc[i] + bv;
    if (act) v = tanhf(v);
    sOut[(i + 8 * half) * outStride + col] = v;   // C/D layout: VGPR i -> row i / i+8
  }
}

__global__ __launch_bounds__(256) void fused_gnn(Params P) {
  __shared__ float sX[16 * XSTR];    // wide concat input buffer
  __shared__ float sH[16 * HSTR];    // node state h
  __shared__ float sT[16 * HSTR];    // layer temp
  __shared__ float sEG[16 * 64];     // gathered ex_x per row
  __shared__ float sRel[16 * 2];

  const int tid = threadIdx.x;
  const int lane = tid & 31;
  const int wave = tid >> 5;
  const int colBase = wave * 16;
  const int r0 = blockIdx.x * 16;

  // ---- stage ex_g (gather 64 features per row from era/pan at nn_idx) ----
  for (int idx = tid; idx < 16 * 64; idx += 256) {
    const int row = idx >> 6, j = idx & 63;
    const int r = r0 + row, b = r / NSTA, n = r % NSTA;
    const int g = P.nn[n], gi = g >> 7, gj = g & 127;
    const int l2 = j >> 2, c = j & 3;
    float v;
    if (l2 < 8) v = P.era[(((b * 4 + c) * 128 + gi) * 129 + gj) * 8 + l2];
    else        v = P.pan[(((b * 4 + c) * 128 + gi) * 128 + gj) * 8 + (l2 - 8)];
    sEG[idx] = v;
  }
  for (int idx = tid; idx < 32; idx += 256) {
    const int row = idx >> 1, k = idx & 1;
    sRel[idx] = P.rel[((r0 + row) % NSTA) * 2 + k];
  }
  // ---- stage embedding input: [madis_x(32) | pos(2) | 0 0] ----
  for (int idx = tid; idx < 16 * 36; idx += 256) {
    const int row = idx / 36, c2 = idx % 36;
    const int r = r0 + row, b = r / NSTA, n = r % NSTA;
    float v;
    if (c2 < 32) { const int l = c2 >> 2, ch = c2 & 3;
                   v = P.obs[b * 768 + ch * 192 + n * 8 + l]; }
    else if (c2 == 32) v = P.csta[n * 2 + 1];
    else if (c2 == 33) v = P.csta[n * 2 + 0];
    else v = 0.0f;
    sX[row * XSTR + c2] = v;
  }
  __syncthreads();

  // ---- embedding MLP: h = tanh(tanh(x@e1+b1)@e2+b2) ----
  mm_layer(P.e1w, P.e1b, sX, 34, XSTR, sT, HSTR, 128, true, lane, colBase);
  __syncthreads();
  mm_layer(P.e2w, P.e2b, sT, 128, HSTR, sH, HSTR, 128, true, lane, colBase);
  __syncthreads();

  // ---- two GNN exchange blocks ----
  for (int ex = 0; ex < 2; ++ex) {
    // stage msg input: [h(128) | ex_g(64) | rel(2) | 0 0]
    for (int idx = tid; idx < 16 * 196; idx += 256) {
      const int row = idx / 196, c2 = idx % 196;
      float v;
      if (c2 < 128)      v = sH[row * HSTR + c2];
      else if (c2 < 192) v = sEG[row * 64 + (c2 - 128)];
      else if (c2 < 194) v = sRel[row * 2 + (c2 - 192)];
      else               v = 0.0f;
      sX[row * XSTR + c2] = v;
    }
    __syncthreads();
    mm_layer(P.m1w[ex], P.m1b[ex], sX, 194, XSTR, sT, HSTR, 128, true, lane, colBase);
    __syncthreads();
    // m -> sX cols 128..255 (upd input becomes [h | m]; h still in cols 0..127)
    mm_layer(P.m2w[ex], P.m2b[ex], sT, 128, HSTR, sX + 128, XSTR, 128, true, lane, colBase);
    __syncthreads();
    mm_layer(P.u1w[ex], P.u1b[ex], sX, 256, XSTR, sT, HSTR, 128, true, lane, colBase);
    __syncthreads();
    mm_layer(P.u2w[ex], P.u2b[ex], sT, 128, HSTR, sH, HSTR, 128, false, lane, colBase);
    __syncthreads();
  }

  // ---- output head: o = tanh(h@o1+b)@o2+b ----
  mm_layer(P.o1w, P.o1b, sH, 128, HSTR, sT, HSTR, 128, true, lane, colBase);
  __syncthreads();

  if (wave < 2) {  // 24 output cols -> 2 column tiles; EXEC all-1s within wave
    const int half = lane >> 4, lr = lane & 15;
    const int col = wave * 16 + lr;
    v8f acc = {};
    for (int k = 0; k < 128; k += 4) {
      const int ka = k + 2 * half;
      v2f a, bw;
      a.x = sT[lr * HSTR + ka];
      a.y = sT[lr * HSTR + ka + 1];
      bw.x = (col < 24) ? P.o2w[ka * 24 + col]       : 0.0f;
      bw.y = (col < 24) ? P.o2w[(ka + 1) * 24 + col] : 0.0f;
      acc = __builtin_amdgcn_wmma_f32_16x16x4_f32(false, a, false, bw,
                                                  (short)0, acc, false, false);
    }
    if (col < 24) {
      const float bv = P.o2b[col];
#pragma unroll
      for (int i = 0; i < 8; ++i) {
        const int r = r0 + i + 8 * half;
        const int b = r / NSTA, n = r % NSTA;
        P.out[b * (NSTA * 24) + n * 24 + col] = acc[i] + bv;   // (B,1,N,24) flat
      }
    }
  }
}

// ---------------------------------------------------------------------------
// Launch wrapper.  d_in order: obs_his, era_his, pan_fut, csta, cera, cpan,
// then params leaves (w,b per linear) in key order:
// emb1, emb2, ex1.{msg1,msg2,upd1,upd2}, ex2.{...}, out1, out2.
// ---------------------------------------------------------------------------
extern "C" void kernel_launch(void* const* d_in, const int* in_sizes, int n_in,
                              void* d_out, int out_size, void* d_ws, size_t ws_size,
                              hipStream_t stream) {
  (void)in_sizes; (void)n_in; (void)out_size; (void)ws_size;
  const float* obs  = (const float*)d_in[0];
  const float* era  = (const float*)d_in[1];
  const float* pan  = (const float*)d_in[2];
  const float* csta = (const float*)d_in[3];
  const float* cera = (const float*)d_in[4];
  // d_in[5] = cpan (unused by the reference forward)

  int*   nn  = (int*)d_ws;
  float* rel = (float*)((char*)d_ws + 32 * sizeof(int));

  nn_kernel<<<NSTA, 256, 0, stream>>>(csta, cera, nn, rel);

  Params P;
  P.obs = obs; P.era = era; P.pan = pan; P.csta = csta;
  int i = 6;
  P.e1w = (const float*)d_in[i++]; P.e1b = (const float*)d_in[i++];
  P.e2w = (const float*)d_in[i++]; P.e2b = (const float*)d_in[i++];
  for (int ex = 0; ex < 2; ++ex) {
    P.m1w[ex] = (const float*)d_in[i++]; P.m1b[ex] = (const float*)d_in[i++];
    P.m2w[ex] = (const float*)d_in[i++]; P.m2b[ex] = (const float*)d_in[i++];
    P.u1w[ex] = (const float*)d_in[i++]; P.u1b[ex] = (const float*)d_in[i++];
    P.u2w[ex] = (const float*)d_in[i++]; P.u2b[ex] = (const float*)d_in[i++];
  }
  P.o1w = (const float*)d_in[i++]; P.o1b = (const float*)d_in[i++];
  P.o2w = (const float*)d_in[i++]; P.o2b = (const float*)d_in[i++];
  P.nn = nn; P.rel = rel; P.out = (float*)d_out;

  fused_gnn<<<TILES, 256, 0, stream>>>(P);
}